// GCN_31903017075238
// MI455X (gfx1250) — compile-verified
//
#include <hip/hip_runtime.h>
#include <math.h>

#define NN   100000
#define EE   1600000
#define GG   1000
#define F_IN 16
#define HH   64
#define CC   2
#define EPSN 1e-5f

typedef __attribute__((ext_vector_type(2))) float v2f;
typedef __attribute__((ext_vector_type(8))) float v8f;

// ---------------- degree / counts ----------------
__global__ void k_counts(const int* __restrict__ batch, float* __restrict__ counts) {
  int i = blockIdx.x * blockDim.x + threadIdx.x;
  if (i < NN) atomicAdd(&counts[batch[i]], 1.0f);
}
__global__ void k_deg(const int* __restrict__ dst, float* __restrict__ deg) {
  int e = blockIdx.x * blockDim.x + threadIdx.x;
  if (e < EE) atomicAdd(&deg[dst[e]], 1.0f);
}
__global__ void k_cinv(const float* __restrict__ counts, float* __restrict__ cinv) {
  int g = blockIdx.x * blockDim.x + threadIdx.x;
  if (g < GG) cinv[g] = 1.0f / fmaxf(counts[g], 1.0f);
}
__global__ void k_dinv(float* __restrict__ d) {  // in-place: deg -> rsqrt(deg+1)
  int i = blockIdx.x * blockDim.x + threadIdx.x;
  if (i < NN) d[i] = rsqrtf(d[i] + 1.0f);
}

// ---------------- GraphNorm ----------------
__global__ void k_gn_mean(const float* __restrict__ x, const int* __restrict__ batch,
                          float* __restrict__ mean, int F, int tot) {
  int t = blockIdx.x * blockDim.x + threadIdx.x;
  if (t >= tot) return;
  int i = t / F, f = t - i * F;
  atomicAdd(&mean[batch[i] * F + f], x[t]);
}
__global__ void k_gn_center_var(const float* __restrict__ xin, float* __restrict__ xout,
                                const int* __restrict__ batch, const float* __restrict__ mean,
                                const float* __restrict__ cinv, const float* __restrict__ alpha,
                                float* __restrict__ var, int F, int tot) {
  int t = blockIdx.x * blockDim.x + threadIdx.x;
  if (t >= tot) return;
  int i = t / F, f = t - i * F;
  int g = batch[i];
  float c = xin[t] - alpha[f] * mean[g * F + f] * cinv[g];
  xout[t] = c;
  atomicAdd(&var[g * F + f], c * c);
}
__global__ void k_gn_final(float* __restrict__ x, const int* __restrict__ batch,
                           const float* __restrict__ var, const float* __restrict__ cinv,
                           const float* __restrict__ w, const float* __restrict__ b,
                           int F, int tot) {
  int t = blockIdx.x * blockDim.x + threadIdx.x;
  if (t >= tot) return;
  int i = t / F, f = t - i * F;
  int g = batch[i];
  x[t] = w[f] * (x[t] * rsqrtf(var[g * F + f] * cinv[g] + EPSN)) + b[f];
}

// ---------------- dense H = X @ W via V_WMMA_F32_16X16X4_F32 ----------------
// X: [NN x K] row-major (K = 16 or 64), W: [K x 64] row-major, H: [NN x 64].
// Block = 128 threads = 4 wave32s; wave w computes 16x16 tile at cols w*16.
__global__ void k_gemm_wmma(const float* __restrict__ A, const float* __restrict__ W,
                            float* __restrict__ H, int K) {
  int wave = threadIdx.x >> 5;
  int lane = threadIdx.x & 31;
  int half = lane >> 4;   // 0: lanes 0-15, 1: lanes 16-31
  int l16  = lane & 15;
  int m0 = blockIdx.x * 16;
  int n0 = wave * 16;
  v8f c = {};
  for (int kk = 0; kk < K; kk += 4) {
    int ka = kk + half * 2;  // ISA 16x4 fp32 A layout: lanes16-31 carry K=2,3
    v2f a, b;
    const float* arow = A + (m0 + l16) * K + ka;
    a.x = arow[0];
    a.y = arow[1];
    const float* bcol = W + ka * HH + n0 + l16;  // B 4x16: rows striped across lanes
    b.x = bcol[0];
    b.y = bcol[HH];
    c = __builtin_amdgcn_wmma_f32_16x16x4_f32(false, a, false, b, (short)0, c,
                                              false, false);
  }
  // D 16x16 fp32: VGPR r -> row r (lanes 0-15) / row r+8 (lanes 16-31)
  #pragma unroll
  for (int r = 0; r < 8; r++) {
    H[(m0 + r + half * 8) * HH + n0 + l16] = c[r];
  }
}

// ---------------- edge scatter: agg[dst] += h[src] * dinv[s]*dinv[d] --------
__global__ void k_scatter(const float* __restrict__ h, const int* __restrict__ src,
                          const int* __restrict__ dst, const float* __restrict__ dinv,
                          float* __restrict__ agg, int tot) {
  int t = blockIdx.x * blockDim.x + threadIdx.x;
  if (t >= tot) return;
  int e  = t >> 4;            // 16 threads per edge
  int f4 = (t & 15) << 2;     // 4 features per thread
  int s = src[e], d = dst[e];
  float coef = dinv[s] * dinv[d];
  const float4 v = *(const float4*)(h + (size_t)s * HH + f4);
  float* base = agg + (size_t)d * HH + f4;
  atomicAdd(base + 0, v.x * coef);
  atomicAdd(base + 1, v.y * coef);
  atomicAdd(base + 2, v.z * coef);
  atomicAdd(base + 3, v.w * coef);
}

// ---------------- self-loop + bias + relu ----------------
__global__ void k_self_bias_relu(float* __restrict__ agg, const float* __restrict__ h,
                                 const float* __restrict__ dinv, const float* __restrict__ b,
                                 int tot) {
  int t = blockIdx.x * blockDim.x + threadIdx.x;
  if (t >= tot) return;
  int i = t >> 6, f = t & 63;
  float di = dinv[i];
  agg[t] = fmaxf(agg[t] + h[t] * di * di + b[f], 0.0f);
}

// ---------------- mean pool accumulate ----------------
__global__ void k_pool(const float* __restrict__ x, const int* __restrict__ batch,
                       float* __restrict__ pooled, int tot) {
  int t = blockIdx.x * blockDim.x + threadIdx.x;
  if (t >= tot) return;
  int i = t >> 6, f = t & 63;
  atomicAdd(&pooled[batch[i] * HH + f], x[t]);
}

// ---------------- MLP head + softmax (one block per graph) ----------------
__global__ void k_mlp(const float* __restrict__ pooled, const float* __restrict__ cinv,
                      const float* __restrict__ Wd, const float* __restrict__ bd,
                      const float* __restrict__ Wo, const float* __restrict__ bo,
                      float* __restrict__ out) {
  int g = blockIdx.x;
  int t = threadIdx.x;  // 64 threads
  __shared__ float p[HH], hs[HH], lg[CC];
  p[t] = pooled[g * HH + t] * cinv[g];
  __syncthreads();
  float acc = bd[t];
  for (int k = 0; k < HH; k++) acc += p[k] * Wd[k * HH + t];
  hs[t] = fmaxf(acc, 0.0f);
  __syncthreads();
  if (t < CC) {
    float s = bo[t];
    for (int k = 0; k < HH; k++) s += hs[k] * Wo[k * CC + t];
    lg[t] = s;
  }
  __syncthreads();
  if (t < CC) {
    float m  = fmaxf(lg[0], lg[1]);
    float e0 = expf(lg[0] - m), e1 = expf(lg[1] - m);
    out[g * CC + t] = (t == 0 ? e0 : e1) / (e0 + e1);
  }
}

// ---------------- driver ----------------
extern "C" void kernel_launch(void* const* d_in, const int* in_sizes, int n_in,
                              void* d_out, int out_size, void* d_ws, size_t ws_size,
                              hipStream_t stream) {
  const float* x_in  = (const float*)d_in[0];
  const int*   edge  = (const int*)d_in[1];
  const int*   srcI  = edge;
  const int*   dstI  = edge + EE;
  const int*   batch = (const int*)d_in[2];
  const float* gn0_w = (const float*)d_in[3];
  const float* gn0_b = (const float*)d_in[4];
  const float* gn0_a = (const float*)d_in[5];
  const float* W1    = (const float*)d_in[6];
  const float* b1    = (const float*)d_in[7];
  const float* gn1_w = (const float*)d_in[8];
  const float* gn1_b = (const float*)d_in[9];
  const float* gn1_a = (const float*)d_in[10];
  const float* W2    = (const float*)d_in[11];
  const float* b2    = (const float*)d_in[12];
  const float* gn2_w = (const float*)d_in[13];
  const float* gn2_b = (const float*)d_in[14];
  const float* gn2_a = (const float*)d_in[15];
  const float* W3    = (const float*)d_in[16];
  const float* b3    = (const float*)d_in[17];
  const float* Wd    = (const float*)d_in[18];
  const float* bd    = (const float*)d_in[19];
  const float* Wo    = (const float*)d_in[20];
  const float* bo    = (const float*)d_in[21];
  float* out = (float*)d_out;

  // carve workspace
  char* ws = (char*)d_ws;
  auto alloc = [&](size_t nfloats) {
    float* p = (float*)ws;
    ws += ((nfloats * sizeof(float) + 255) / 256) * 256;
    return p;
  };
  float* buf0   = alloc((size_t)NN * HH);  // x / agg
  float* buf1   = alloc((size_t)NN * HH);  // h = x @ W
  float* counts = alloc(GG);
  float* cinv   = alloc(GG);
  float* dinv   = alloc(NN);               // deg accumulator, then rsqrt(deg+1)
  float* mean   = alloc((size_t)GG * HH);
  float* var    = alloc((size_t)GG * HH);
  float* pooled = alloc((size_t)GG * HH);

  const int TB = 256;
  auto nb = [](int n, int b) { return (n + b - 1) / b; };

  // counts / degrees
  hipMemsetAsync(counts, 0, GG * sizeof(float), stream);
  hipMemsetAsync(dinv, 0, NN * sizeof(float), stream);
  k_counts<<<nb(NN, TB), TB, 0, stream>>>(batch, counts);
  k_deg<<<nb(EE, TB), TB, 0, stream>>>(dstI, dinv);
  k_cinv<<<nb(GG, TB), TB, 0, stream>>>(counts, cinv);
  k_dinv<<<nb(NN, TB), TB, 0, stream>>>(dinv);

  auto graphnorm = [&](const float* xin, float* xio, int F, const float* w,
                       const float* b, const float* a) {
    hipMemsetAsync(mean, 0, (size_t)GG * F * sizeof(float), stream);
    hipMemsetAsync(var, 0, (size_t)GG * F * sizeof(float), stream);
    int tot = NN * F;
    k_gn_mean<<<nb(tot, TB), TB, 0, stream>>>(xin, batch, mean, F, tot);
    k_gn_center_var<<<nb(tot, TB), TB, 0, stream>>>(xin, xio, batch, mean, cinv, a,
                                                    var, F, tot);
    k_gn_final<<<nb(tot, TB), TB, 0, stream>>>(xio, batch, var, cinv, w, b, F, tot);
  };

  auto gcn_conv = [&](float* xcur /*buf0*/, float* h /*buf1*/, int K,
                      const float* W, const float* bias) {
    k_gemm_wmma<<<NN / 16, 128, 0, stream>>>(xcur, W, h, K);
    hipMemsetAsync(xcur, 0, (size_t)NN * HH * sizeof(float), stream);  // agg
    int e16 = EE * 16;
    k_scatter<<<nb(e16, TB), TB, 0, stream>>>(h, srcI, dstI, dinv, xcur, e16);
    int tot = NN * HH;
    k_self_bias_relu<<<nb(tot, TB), TB, 0, stream>>>(xcur, h, dinv, bias, tot);
  };

  // layer 0 norm: d_in x -> buf0 (N x 16)
  graphnorm(x_in, buf0, F_IN, gn0_w, gn0_b, gn0_a);
  // conv1 (16 -> 64) + relu, result in buf0
  gcn_conv(buf0, buf1, F_IN, W1, b1);
  // norm1 in place
  graphnorm(buf0, buf0, HH, gn1_w, gn1_b, gn1_a);
  // conv2 + relu
  gcn_conv(buf0, buf1, HH, W2, b2);
  // norm2 in place
  graphnorm(buf0, buf0, HH, gn2_w, gn2_b, gn2_a);
  // conv3 + relu
  gcn_conv(buf0, buf1, HH, W3, b3);

  // global mean pool
  hipMemsetAsync(pooled, 0, (size_t)GG * HH * sizeof(float), stream);
  int tot = NN * HH;
  k_pool<<<nb(tot, TB), TB, 0, stream>>>(buf0, batch, pooled, tot);

  // MLP head + softmax
  k_mlp<<<GG, HH, 0, stream>>>(pooled, cinv, Wd, bd, Wo, bo, out);
}